// HydrogelGNNPINN_84696755077245
// MI455X (gfx1250) — compile-verified
//
#include <hip/hip_runtime.h>

typedef __attribute__((ext_vector_type(2))) float v2f;
typedef __attribute__((ext_vector_type(8))) float v8f;

#define DFEAT 64

// ---------------------------------------------------------------- degree pass
__global__ void k_set1(float* __restrict__ p, int n) {
    int i = blockIdx.x * blockDim.x + threadIdx.x;
    if (i < n) p[i] = 1.0f;  // self-loop contributes 1 to every node's degree
}

__global__ void k_deg(const long long* __restrict__ dst, float* __restrict__ deg, int e) {
    int i = blockIdx.x * blockDim.x + threadIdx.x;
    if (i < e) unsafeAtomicAdd(&deg[(int)dst[i]], 1.0f);
}

__global__ void k_rsqrt(float* __restrict__ p, int n) {
    int i = blockIdx.x * blockDim.x + threadIdx.x;
    if (i < n) p[i] = rsqrtf(p[i]);  // deg >= 1 always (self-loops)
}

// ------------------------------------------------- H = (relu?)(A) @ W  [NxD * DxD]
// One wave per 16x16 output tile; 8 waves/block = 2 row-tiles x 4 col-tiles.
// fp32 WMMA 16x16x4: A-frag lane L -> M=L%16, K = kb + 2*(L/16) + {0,1} (float2 load)
//                    B-frag lane L -> N=L%16, same K mapping
//                    C/D  lane L, vgpr v -> M = v + 8*(L/16), N = L%16
template <bool RELU>
__global__ void __launch_bounds__(256) k_gemm64(const float* __restrict__ A,
                                                const float* __restrict__ W,
                                                float* __restrict__ O) {
    int tid  = threadIdx.x;
    int w    = tid >> 5;
    int lane = tid & 31;
    int rt   = blockIdx.x * 2 + (w >> 2);
    int ct   = w & 3;
    int row0 = rt * 16, col0 = ct * 16;
    int m  = lane & 15;      // A row / B col within tile
    int hi = lane >> 4;      // K-half selector

    v8f c = {};
    const float* arow = A + (size_t)(row0 + m) * DFEAT + 2 * hi;
    const float* wcol = W + (size_t)(2 * hi) * DFEAT + col0 + m;

#pragma unroll
    for (int kb = 0; kb < DFEAT; kb += 4) {
        v2f a = *(const v2f*)(arow + kb);
        if (RELU) { a.x = fmaxf(a.x, 0.0f); a.y = fmaxf(a.y, 0.0f); }
        v2f b;
        b.x = wcol[kb * DFEAT];
        b.y = wcol[kb * DFEAT + DFEAT];
        c = __builtin_amdgcn_wmma_f32_16x16x4_f32(false, a, false, b,
                                                  (short)0, c, false, false);
    }

    int hi8 = 8 * hi;
#pragma unroll
    for (int v = 0; v < 8; ++v)
        O[(size_t)(row0 + v + hi8) * DFEAT + col0 + m] = c[v];
}

// ------------------------------------------- agg init: self-loop term + bias
__global__ void k_agg_init(float* __restrict__ out, const float* __restrict__ h,
                           const float* __restrict__ dinv, const float* __restrict__ bias,
                           int n) {
    int idx = blockIdx.x * blockDim.x + threadIdx.x;
    if (idx >= n * DFEAT) return;
    int node = idx >> 6;
    int f    = idx & 63;
    float di = dinv[node];
    out[idx] = h[idx] * di * di + bias[f];
}

// ------------------------------------- agg edges: scatter norm * h[src] -> dst
__global__ void k_agg_edges(float* __restrict__ out, const float* __restrict__ h,
                            const float* __restrict__ dinv,
                            const long long* __restrict__ src,
                            const long long* __restrict__ dst, int e) {
    int t = blockIdx.x * blockDim.x + threadIdx.x;
    if (t >= e * 16) return;
    int ed = t >> 4;
    int g  = t & 15;
    int s  = (int)src[ed];
    int d  = (int)dst[ed];
    float wgt = dinv[s] * dinv[d];
    const float4 hv = *(const float4*)(h + (size_t)s * DFEAT + g * 4);
    float* o = out + (size_t)d * DFEAT + g * 4;
    unsafeAtomicAdd(o + 0, wgt * hv.x);
    unsafeAtomicAdd(o + 1, wgt * hv.y);
    unsafeAtomicAdd(o + 2, wgt * hv.z);
    unsafeAtomicAdd(o + 3, wgt * hv.w);
}

// ---------------- fused MLP head: out = relu([gnn|x] @ Wp1 + bp1) @ Wp2 + bp2
// One block (16 waves) per 16-node tile. Each wave computes two 16x16 tiles of
// h2 (K=128 via WMMA, A-frag from gnn for k<64, x for k>=64), relu+bias, stages
// to LDS, contracts with Wp2[512x3] into a shared 16x3 accumulator.
__global__ void __launch_bounds__(512) k_pinn(const float* __restrict__ gnn,
                                              const float* __restrict__ x,
                                              const float* __restrict__ Wp1,
                                              const float* __restrict__ bp1,
                                              const float* __restrict__ Wp2,
                                              const float* __restrict__ bp2,
                                              float* __restrict__ out, int n) {
    __shared__ float s_h2[16][16][17];  // [wave][m][n], padded
    __shared__ float s_acc[16][3];

    int tid  = threadIdx.x;
    int w    = tid >> 5;
    int lane = tid & 31;
    int m    = lane & 15;
    int hi   = lane >> 4;
    int node0 = blockIdx.x * 16;

    if (tid < 48) s_acc[tid / 3][tid % 3] = 0.0f;
    __syncthreads();

    const float* g_a = gnn + (size_t)(node0 + m) * DFEAT + 2 * hi;
    const float* x_a = x   + (size_t)(node0 + m) * DFEAT + 2 * hi;

    for (int cc = 0; cc < 2; ++cc) {
        int ct   = 2 * w + cc;
        int col0 = ct * 16;
        const float* wb = Wp1 + (size_t)(2 * hi) * 512 + col0 + m;

        v8f c = {};
#pragma unroll
        for (int kb = 0; kb < 128; kb += 4) {
            v2f a = (kb < 64) ? *(const v2f*)(g_a + kb)
                              : *(const v2f*)(x_a + (kb - 64));
            v2f b;
            b.x = wb[kb * 512];
            b.y = wb[kb * 512 + 512];
            c = __builtin_amdgcn_wmma_f32_16x16x4_f32(false, a, false, b,
                                                      (short)0, c, false, false);
        }

        float bi  = bp1[col0 + m];
        int   hi8 = 8 * hi;
#pragma unroll
        for (int v = 0; v < 8; ++v)
            s_h2[w][v + hi8][m] = fmaxf(c[v] + bi, 0.0f);
        // same-wave DS ordering: stores visible to our own reads below

        for (int idx = lane; idx < 48; idx += 32) {
            int mm = idx / 3;
            int p  = idx - mm * 3;
            float s = 0.0f;
#pragma unroll
            for (int nn = 0; nn < 16; ++nn)
                s += s_h2[w][mm][nn] * Wp2[(col0 + nn) * 3 + p];
            atomicAdd(&s_acc[mm][p], s);
        }
    }
    __syncthreads();

    if (tid < 48) {
        int mm = tid / 3;
        int p  = tid - mm * 3;
        out[(size_t)(node0 + mm) * 3 + p] = s_acc[mm][p] + bp2[p];
    }
    (void)n;
}

// -------------------------------------------------------------------- launch
extern "C" void kernel_launch(void* const* d_in, const int* in_sizes, int n_in,
                              void* d_out, int out_size, void* d_ws, size_t ws_size,
                              hipStream_t stream) {
    const float*     x   = (const float*)d_in[0];
    const long long* ei  = (const long long*)d_in[1];  // int64 edge_index [2,E]
    const float*     W1  = (const float*)d_in[2];
    const float*     b1  = (const float*)d_in[3];
    const float*     W2  = (const float*)d_in[4];
    const float*     b2  = (const float*)d_in[5];
    const float*     Wp1 = (const float*)d_in[6];
    const float*     bp1 = (const float*)d_in[7];
    const float*     Wp2 = (const float*)d_in[8];
    const float*     bp2 = (const float*)d_in[9];
    float*           out = (float*)d_out;

    int n = in_sizes[0] / DFEAT;   // 100000
    int e = in_sizes[1] / 2;       // 1200000
    const long long* src = ei;
    const long long* dst = ei + e;

    float* ws   = (float*)d_ws;
    float* dinv = ws;
    float* bufA = ws + (((size_t)n + 255) & ~(size_t)255);  // h1, then h2pre
    float* bufB = bufA + (size_t)n * DFEAT;                 // agg1, then gnn_out

    const int T = 256;
    // degree -> dinv
    k_set1 <<<(n + T - 1) / T, T, 0, stream>>>(dinv, n);
    k_deg  <<<(e + T - 1) / T, T, 0, stream>>>(dst, dinv, e);
    k_rsqrt<<<(n + T - 1) / T, T, 0, stream>>>(dinv, n);

    // conv1: h1 = x @ W1 ; agg1 = norm-aggregate(h1) + b1
    k_gemm64<false><<<n / 32, T, 0, stream>>>(x, W1, bufA);
    k_agg_init <<<(n * DFEAT + T - 1) / T, T, 0, stream>>>(bufB, bufA, dinv, b1, n);
    k_agg_edges<<<(e * 16 + T - 1) / T, T, 0, stream>>>(bufB, bufA, dinv, src, dst, e);

    // conv2: h2pre = relu(agg1) @ W2 ; gnn_out = norm-aggregate(h2pre) + b2
    k_gemm64<true><<<n / 32, T, 0, stream>>>(bufB, W2, bufA);
    k_agg_init <<<(n * DFEAT + T - 1) / T, T, 0, stream>>>(bufB, bufA, dinv, b2, n);
    k_agg_edges<<<(e * 16 + T - 1) / T, T, 0, stream>>>(bufB, bufA, dinv, src, dst, e);

    // fused MLP head
    k_pinn<<<n / 16, 512, 0, stream>>>(bufB, x, Wp1, bp1, Wp2, bp2, out, n);

    (void)n_in; (void)out_size; (void)ws_size;
}